// YOLOLossV3_22505628631665
// MI455X (gfx1250) — compile-verified
//
#include <hip/hip_runtime.h>
#include <hip/hip_bf16.h>

// ---------------------------------------------------------------------------
// YOLOv3 loss for MI455X (gfx1250, wave32).
// Strategy: the loss is a sparse reduction. Only the conf channel (1 MB) needs
// a full scan; the 1600 target cells are gathered. All cross-lane sums use
// V_WMMA_F32_16X16X4_F32 as an exact f32 reducer (B = ones so the result does
// not depend on B's K<->VGPR mapping; A/C/D layouts are the documented ones).
// Deterministic two-stage reduction (no float atomics) for graph replay.
// ---------------------------------------------------------------------------

typedef float v2f __attribute__((ext_vector_type(2)));
typedef float v8f __attribute__((ext_vector_type(8)));

#define BSZ    32
#define NANC   3
#define GH     52
#define GW     52
#define HW     (GH * GW)            // 2704
#define NT     50
#define NCLS   80
#define BBOX   85                   // 5 + 80
#define NCELL  (BSZ * NANC * HW)    // 259584 == 1014 * 256 exactly
#define NTARG  (BSZ * NT)           // 1600
#define NBLK_C (NCELL / 256)        // 1014
#define NBLK_D ((NTARG + 255) / 256) // 7

__device__ __forceinline__ float clip01(float p) {
    return fminf(fmaxf(p, 1e-7f), 1.0f - 1e-7f);
}
__device__ __forceinline__ float sigm(float x) {
    return 1.0f / (1.0f + expf(-x));
}
// general BCE matching the reference: p clipped, t arbitrary in [0,1]
__device__ __forceinline__ float bce(float p, float t) {
    p = clip01(p);
    return -(t * logf(p) + (1.0f - t) * logf(1.0f - p));
}

// ---------------------------------------------------------------------------
// Wave32 sum via two V_WMMA_F32_16X16X4_F32 (exact f32, broadcast to all lanes)
// WMMA1: A[m][0]=v(lane m), A[m][2]=v(lane m+16), others 0; B=ones; C=0
//        -> D[m][n] = v[m] + v[m+16].  D layout: lane<16 holds rows 0..7,
//        lane>=16 holds rows 8..15 across the 8 D VGPRs.
// In-lane add of 8 D regs -> half-sums S0 (lanes 0-15) / S1 (lanes 16-31).
// WMMA2 on the half-sums -> S0 + S1 = full wave sum, broadcast everywhere.
// ---------------------------------------------------------------------------
__device__ __forceinline__ float wave_sum_wmma(float v) {
    v2f a;   a[0] = v;    a[1] = 0.0f;
    v2f one; one[0] = 1.0f; one[1] = 1.0f;
    v8f z = {};
    v8f d = __builtin_amdgcn_wmma_f32_16x16x4_f32(false, a, false, one,
                                                  (short)0, z, false, false);
    float s = ((d[0] + d[1]) + (d[2] + d[3])) + ((d[4] + d[5]) + (d[6] + d[7]));
    v2f a2; a2[0] = s; a2[1] = 0.0f;
    v8f d2 = __builtin_amdgcn_wmma_f32_16x16x4_f32(false, a2, false, one,
                                                   (short)0, z, false, false);
    return d2[0];
}

// ---------------------------------------------------------------------------
// Kernel C: scan the conf channel of every cell assuming noobj==1.
// Accumulates  sum_cells BCE(sigmoid(conf), 0)  -> one partial per block.
// ---------------------------------------------------------------------------
__global__ void __launch_bounds__(256)
yolo_conf_scan(const float* __restrict__ in, float* __restrict__ partC) {
    const int c = blockIdx.x * 256 + threadIdx.x;     // exact cover of NCELL
    const int b = c / (NANC * HW);
    const int r = c - b * (NANC * HW);
    const int a = r / HW;
    const int s = r - a * HW;
    const long off = ((long)(b * (NANC * BBOX) + a * BBOX + 4)) * HW + s;
    const float p = clip01(sigm(in[off]));
    const float v = -logf(1.0f - p);                   // BCE(p, 0)

    const float wsum = wave_sum_wmma(v);               // uniform flow, EXEC all 1s
    __shared__ float lds[8];
    const int lane = threadIdx.x & 31, wv = threadIdx.x >> 5;
    if (lane == 0) lds[wv] = wsum;
    __syncthreads();
    if (threadIdx.x == 0) {
        float t = 0.0f;
        for (int i = 0; i < 8; ++i) t += lds[i];
        partC[blockIdx.x] = t;
    }
}

// ---------------------------------------------------------------------------
// Kernel D: one thread per (b, t) target. Builds the target inline (anchor IoU
// argmax, tx/ty/tw/th, class), gathers the 85 channels of the masked cell plus
// the ignored-anchor conf cells, and accumulates 8 partial sums:
//   [0]=sum BCE(x,tx)  [1]=sum BCE(y,ty)  [2]=sum (w-tw)^2  [3]=sum (h-th)^2
//   [4]=sum -log(conf) (t=1 term)         [5]=sum class BCE
//   [6]=sum (BCE(conf,0) - C0) over iou>thr cells (noobj correction)
//   [7]=n_pos
// ---------------------------------------------------------------------------
__global__ void __launch_bounds__(256)
yolo_target_gather(const float* __restrict__ in, const float* __restrict__ tg,
                   float* __restrict__ partD) {
    const float AW[3] = {1.25f, 2.0f, 4.125f};   // ANCHORS / (416/52)
    const float AH[3] = {1.625f, 3.75f, 2.875f};
    const int t = blockIdx.x * 256 + threadIdx.x;

    float acc0 = 0, acc1 = 0, acc2 = 0, acc3 = 0, acc4 = 0, acc5 = 0, acc6 = 0, acc7 = 0;
    if (t < NTARG) {
        const int b = t / NT;
        const float* tr = tg + (long)t * 6;
        const float gx = tr[0] * (float)GW, gy = tr[1] * (float)GH;
        const float gw = tr[2] * (float)GW, gh = tr[3] * (float)GH;
        const int   cls = (int)tr[4];
        const bool  valid = (tr[5] >= 1.0f) && (gw > 0.0f) && (gh > 0.0f);
        const int gi = (int)floorf(gx), gj = (int)floorf(gy);
        if (valid && gi >= 0 && gi < GW && gj >= 0 && gj < GH && cls >= 0 && cls < NCLS) {
            float iou[3];
            for (int a = 0; a < 3; ++a) {
                const float inter = fminf(gw, AW[a]) * fminf(gh, AH[a]);
                const float uni   = gw * gh + AW[a] * AH[a] - inter + 1e-16f;
                iou[a] = inter / uni;
            }
            int best = 0;
            if (iou[1] > iou[best]) best = 1;
            if (iou[2] > iou[best]) best = 2;

            const float tx = gx - floorf(gx), ty = gy - floorf(gy);
            const float tw = logf(gw / AW[best] + 1e-16f);
            const float th = logf(gh / AH[best] + 1e-16f);

            const int  sp = gj * GW + gi;
            const long cb = ((long)(b * (NANC * BBOX) + best * BBOX)) * HW + sp;

            const float px = sigm(in[cb + 0 * HW]);
            const float py = sigm(in[cb + 1 * HW]);
            const float pw = in[cb + 2 * HW];
            const float ph = in[cb + 3 * HW];
            const float pc = sigm(in[cb + 4 * HW]);

            acc0 = bce(px, tx);
            acc1 = bce(py, ty);
            acc2 = (pw - tw) * (pw - tw);
            acc3 = (ph - th) * (ph - th);
            acc4 = -logf(clip01(pc));                   // BCE(conf, 1)

            float sc = 0.0f;
            for (int c2 = 0; c2 < NCLS; ++c2) {
                const float p = clip01(sigm(in[cb + (long)(5 + c2) * HW]));
                sc += (c2 == cls) ? -logf(p) : -logf(1.0f - p);
            }
            acc5 = sc;

            const float C0 = -logf(1.0f - 1e-7f);       // BCE(0,0)
            float s2c = 0.0f;
            for (int a = 0; a < 3; ++a) {
                if (iou[a] > 0.5f) {                    // IGNORE_THR
                    const long co = ((long)(b * (NANC * BBOX) + a * BBOX + 4)) * HW + sp;
                    const float q = clip01(sigm(in[co]));
                    s2c += (-logf(1.0f - q)) - C0;
                }
            }
            acc6 = s2c;
            acc7 = 1.0f;
        }
    }
    // reconverged: EXEC is all-ones here -> WMMA legal
    float acc[8] = {acc0, acc1, acc2, acc3, acc4, acc5, acc6, acc7};
    __shared__ float lds[8 * 8];
    const int lane = threadIdx.x & 31, wv = threadIdx.x >> 5;
    for (int q = 0; q < 8; ++q) {
        const float wsum = wave_sum_wmma(acc[q]);
        if (lane == 0) lds[wv * 8 + q] = wsum;
        __syncthreads();   // keep flow uniform between WMMA calls
    }
    if (threadIdx.x == 0) {
        for (int q = 0; q < 8; ++q) {
            float tsum = 0.0f;
            for (int w2 = 0; w2 < 8; ++w2) tsum += lds[w2 * 8 + q];
            partD[blockIdx.x * 8 + q] = tsum;
        }
    }
}

// ---------------------------------------------------------------------------
// Kernel F: single wave folds all partials and emits the 7 losses.
// ---------------------------------------------------------------------------
__global__ void __launch_bounds__(32)
yolo_finalize(const float* __restrict__ partC, const float* __restrict__ partD,
              float* __restrict__ out) {
    const int lane = threadIdx.x;
    float acc = 0.0f;
    for (int i = lane; i < NBLK_C; i += 32) acc += partC[i];
    const float S2ALL = wave_sum_wmma(acc);            // uniform, full wave

    __shared__ float sums[8];
    if (lane < 8) {
        float a2 = 0.0f;
        for (int b2 = 0; b2 < NBLK_D; ++b2) a2 += partD[b2 * 8 + lane];
        sums[lane] = a2;
    }
    __syncthreads();
    if (lane == 0) {
        const float SX = sums[0], SY = sums[1], SW = sums[2], SH = sums[3];
        const float SC1 = sums[4], SCLS = sums[5], S2C = sums[6], NPOS = sums[7];
        const float Nf = (float)NCELL;
        const float C0 = -logf(1.0f - 1e-7f);

        const float lx = ((Nf - NPOS) * C0 + SX) / Nf;
        const float ly = ((Nf - NPOS) * C0 + SY) / Nf;
        const float lw = SW / Nf;
        const float lh = SH / Nf;
        const float S1 = (Nf - NPOS) * C0 + SC1;       // BCE(conf*mask, mask) sum
        const float S2 = S2ALL - S2C;                  // BCE(conf*noobj, 0) sum
        const float lconf = S1 / Nf + 0.5f * (S2 / Nf);
        const float np = fmaxf(NPOS, 1.0f);
        const float lcls = SCLS / (np * (float)NCLS);
        const float loss = (lx + ly) * 2.5f + (lw + lh) * 2.5f + lconf + lcls;

        out[0] = loss; out[1] = lx; out[2] = ly; out[3] = lw;
        out[4] = lh;   out[5] = lconf; out[6] = lcls;
    }
}

extern "C" void kernel_launch(void* const* d_in, const int* in_sizes, int n_in,
                              void* d_out, int out_size, void* d_ws, size_t ws_size,
                              hipStream_t stream) {
    (void)in_sizes; (void)n_in; (void)out_size; (void)ws_size;
    const float* in = (const float*)d_in[0];   // (32, 255, 52, 52) f32
    const float* tg = (const float*)d_in[1];   // (32, 50, 6)       f32
    float* out = (float*)d_out;                // 7 f32
    float* ws  = (float*)d_ws;
    float* partC = ws;            // NBLK_C floats (fully written before read)
    float* partD = ws + 2048;     // NBLK_D*8 floats

    yolo_conf_scan    <<<NBLK_C, 256, 0, stream>>>(in, partC);
    yolo_target_gather<<<NBLK_D, 256, 0, stream>>>(in, tg, partD);
    yolo_finalize     <<<1,      32,  0, stream>>>(partC, partD, out);
}